// LSTM_35725537968471
// MI455X (gfx1250) — compile-verified
//
#include <hip/hip_runtime.h>
#include <math.h>

typedef float v2f __attribute__((ext_vector_type(2)));
typedef float v8f __attribute__((ext_vector_type(8)));

#define SEQL 512
#define BSZ  64
#define HD   1024
#define N4   4096          // 4*HD
#define NSEG 4             // K-split of the recurrent GEMM
#define KSEG (HD / NSEG)   // 256

#define GEMM_CHUNK    128
#define GEMM_LDSTRIDE 132  // 128 + pad: 16B-aligned rows, conflict-free b64 reads
#define STEP_LDSTRIDE 260  // 256 + pad: 16B-aligned rows, conflict-free b64 reads

__device__ __forceinline__ float sig_(float x) { return 1.0f / (1.0f + __expf(-x)); }

// ---------------------------------------------------------------------------
__global__ void zero_f32(float* __restrict__ p, int n) {
    int i = blockIdx.x * blockDim.x + threadIdx.x;
    if (i < n) p[i] = 0.0f;
}

__global__ void copy_f32(float* __restrict__ dst, const float* __restrict__ src, int n) {
    int i = blockIdx.x * blockDim.x + threadIdx.x;
    if (i < n) dst[i] = src[i];
}

// ---------------------------------------------------------------------------
// C[M][N4] = A[M][K] @ B[K][N4] + (b1[n] + b2[n])
// Block = 8 waves sharing one 16-row A slab (staged via LDS); each wave owns a
// 16x64 output strip (4 fp32 WMMA accumulators along N).
__global__ __launch_bounds__(256) void gemm_bias_wmma(
    const float* __restrict__ A, const float* __restrict__ B,
    const float* __restrict__ b1, const float* __restrict__ b2,
    float* __restrict__ C, int K)
{
    __shared__ __align__(16) float lds[16 * GEMM_LDSTRIDE];
    const int tid  = threadIdx.x;
    const int lane = tid & 31;
    const int wid  = tid >> 5;
    const int gw   = blockIdx.x * 8 + wid;
    const int m0   = (gw / (N4 / 64)) * 16;   // uniform across the block
    const int n0   = (gw % (N4 / 64)) * 64;
    const int half = lane >> 4;
    const int lm   = lane & 15;

    v8f acc[4];
#pragma unroll
    for (int g = 0; g < 4; ++g) {
        float bv = b1[n0 + g * 16 + lm] + b2[n0 + g * 16 + lm];
#pragma unroll
        for (int r = 0; r < 8; ++r) acc[g][r] = bv;
    }

    for (int k0 = 0; k0 < K; k0 += GEMM_CHUNK) {
        __syncthreads();
        // cooperative stage of A[m0..m0+15][k0..k0+127]: 512 float4, 2/thread
#pragma unroll
        for (int t = 0; t < 2; ++t) {
            int f   = tid + t * 256;
            int row = f >> 5;            // 32 float4 per row
            int c4  = (f & 31) << 2;
            float4 v = *(const float4*)(A + (size_t)(m0 + row) * K + k0 + c4);
            *(float4*)&lds[row * GEMM_LDSTRIDE + c4] = v;
        }
        __syncthreads();

        for (int kk = 0; kk < GEMM_CHUNK; kk += 4) {
            v2f a;
            a.x = lds[lm * GEMM_LDSTRIDE + kk + 2 * half];
            a.y = lds[lm * GEMM_LDSTRIDE + kk + 2 * half + 1];
            const float* brow0 = B + (size_t)(k0 + kk + 2 * half) * N4 + n0 + lm;
            const float* brow1 = brow0 + N4;
#pragma unroll
            for (int g = 0; g < 4; ++g) {
                v2f b;
                b.x = brow0[g * 16];
                b.y = brow1[g * 16];
                acc[g] = __builtin_amdgcn_wmma_f32_16x16x4_f32(
                    false, a, false, b, (short)0, acc[g], false, false);
            }
        }
    }

#pragma unroll
    for (int g = 0; g < 4; ++g)
#pragma unroll
        for (int r = 0; r < 8; ++r) {
            int m = m0 + r + 8 * half;
            int n = n0 + g * 16 + lm;
            C[(size_t)m * N4 + n] = acc[g][r];
        }
}

// ---------------------------------------------------------------------------
// Recurrent-step partial GEMM: part[seg] = h_prev[:, seg*KSEG:(seg+1)*KSEG]
//                                          @ W_hh[seg*KSEG:(seg+1)*KSEG, :]
// Grid: 2 m-groups x NSEG k-segments x 8 j-blocks = 64 blocks of 8 waves.
// Each wave: 32 batch rows (2 WMMA m-tiles) x 16 hidden cols x 4 gates;
// B elements are loaded once and feed two WMMAs. h slab staged in LDS.
__global__ __launch_bounds__(256) void lstm_step_partial(
    const float* __restrict__ Whh,     // [HD][N4]
    const float* __restrict__ hprev,   // [BSZ][HD]
    float* __restrict__ part)          // [NSEG][BSZ][N4]
{
    __shared__ __align__(16) float lds[32 * STEP_LDSTRIDE];
    const int tid  = threadIdx.x;
    const int lane = tid & 31;
    const int wid  = tid >> 5;
    const int jb   = blockIdx.x & 7;
    const int seg  = (blockIdx.x >> 3) & (NSEG - 1);
    const int mgrp = blockIdx.x >> 5;          // 0..1
    const int j0   = (jb * 8 + wid) * 16;
    const int half = lane >> 4;
    const int lm   = lane & 15;
    const int k0   = seg * KSEG;

    // cooperative stage of hprev[mgrp*32 .. +31][k0 .. k0+255]: 2048 float4
#pragma unroll
    for (int t = 0; t < 8; ++t) {
        int f   = tid + t * 256;
        int row = f >> 6;                // 64 float4 per row
        int c4  = (f & 63) << 2;
        float4 v = *(const float4*)(hprev + (size_t)(mgrp * 32 + row) * HD + k0 + c4);
        *(float4*)&lds[row * STEP_LDSTRIDE + c4] = v;
    }
    __syncthreads();

    v8f acc[2][4];
#pragma unroll
    for (int mt = 0; mt < 2; ++mt)
#pragma unroll
        for (int g = 0; g < 4; ++g)
#pragma unroll
            for (int r = 0; r < 8; ++r) acc[mt][g][r] = 0.0f;

    for (int kk = 0; kk < KSEG; kk += 4) {
        v2f a0, a1;
        a0.x = lds[lm * STEP_LDSTRIDE + kk + 2 * half];
        a0.y = lds[lm * STEP_LDSTRIDE + kk + 2 * half + 1];
        a1.x = lds[(16 + lm) * STEP_LDSTRIDE + kk + 2 * half];
        a1.y = lds[(16 + lm) * STEP_LDSTRIDE + kk + 2 * half + 1];
        const float* brow0 = Whh + (size_t)(k0 + kk + 2 * half) * N4 + j0 + lm;
        const float* brow1 = brow0 + N4;
#pragma unroll
        for (int g = 0; g < 4; ++g) {
            v2f b;
            b.x = brow0[g * HD];
            b.y = brow1[g * HD];
            acc[0][g] = __builtin_amdgcn_wmma_f32_16x16x4_f32(
                false, a0, false, b, (short)0, acc[0][g], false, false);
            acc[1][g] = __builtin_amdgcn_wmma_f32_16x16x4_f32(
                false, a1, false, b, (short)0, acc[1][g], false, false);
        }
    }

    float* po = part + (size_t)seg * BSZ * N4;
#pragma unroll
    for (int mt = 0; mt < 2; ++mt)
#pragma unroll
        for (int g = 0; g < 4; ++g)
#pragma unroll
            for (int r = 0; r < 8; ++r) {
                int m = mgrp * 32 + mt * 16 + r + 8 * half;
                po[(size_t)m * N4 + g * HD + j0 + lm] = acc[mt][g][r];
            }
}

// ---------------------------------------------------------------------------
// Segment-sum + LSTM gate fusion: ifgo = xproj_t + sum(part[seg]); then
// c' = sig(f)*c + sig(i)*tanh(g); h' = sig(o)*tanh(c'). One thread per (m,j).
__global__ __launch_bounds__(256) void lstm_gate_reduce(
    const float* __restrict__ xproj_t,  // [BSZ][N4]
    const float* __restrict__ part,     // [NSEG][BSZ][N4]
    float* __restrict__ hnext,          // [BSZ][HD]
    float* __restrict__ cstate,         // [BSZ][HD] in/out
    float* __restrict__ yout)           // [BSZ][HD]
{
    int i = blockIdx.x * blockDim.x + threadIdx.x;   // 0..BSZ*HD-1
    int m = i >> 10;          // / HD
    int j = i & (HD - 1);
    float g4[4];
#pragma unroll
    for (int g = 0; g < 4; ++g) {
        size_t base = (size_t)m * N4 + g * HD + j;
        float s = xproj_t[base];
#pragma unroll
        for (int sgi = 0; sgi < NSEG; ++sgi) s += part[(size_t)sgi * BSZ * N4 + base];
        g4[g] = s;
    }
    float iv = sig_(g4[0]);
    float fv = sig_(g4[1]);
    float gv = tanhf(g4[2]);
    float ov = sig_(g4[3]);
    float cn = fv * cstate[i] + iv * gv;
    float hn = ov * tanhf(cn);
    cstate[i] = cn;
    hnext[i]  = hn;
    yout[i]   = hn;
}

// ---------------------------------------------------------------------------
extern "C" void kernel_launch(void* const* d_in, const int* in_sizes, int n_in,
                              void* d_out, int out_size, void* d_ws, size_t ws_size,
                              hipStream_t stream) {
    const float* x    = (const float*)d_in[0];
    const float* Wih0 = (const float*)d_in[1];
    const float* Whh0 = (const float*)d_in[2];
    const float* bih0 = (const float*)d_in[3];
    const float* bhh0 = (const float*)d_in[4];
    const float* Wih1 = (const float*)d_in[5];
    const float* Whh1 = (const float*)d_in[6];
    const float* bih1 = (const float*)d_in[7];
    const float* bhh1 = (const float*)d_in[8];
    float* out = (float*)d_out;

    const size_t SBH = (size_t)SEQL * BSZ * HD;
    const size_t BH  = (size_t)BSZ * HD;
    float* xproj = (float*)d_ws;                    // SEQL*BSZ*N4
    float* y0    = xproj + (size_t)SEQL * BSZ * N4; // SEQL*BSZ*HD
    float* hbuf  = y0 + SBH;                        // 2*BH ping-pong
    float* cbuf  = hbuf + 2 * BH;                   // BH
    float* part  = cbuf + BH;                       // NSEG*BSZ*N4

    float* y1 = out;                 // (SEQL, BSZ, HD)
    float* hn = out + SBH;           // (2, BSZ, HD)
    float* cn = hn + 2 * BH;         // (2, BSZ, HD)

    const int M = SEQL * BSZ;                          // 32768
    dim3 blk(256);
    dim3 gemmGrid((M / 16) * (N4 / 64) / 8);           // 16384 blocks
    dim3 stepGrid(2 * NSEG * 8);                       // 64 blocks
    dim3 gateGrid((unsigned)(BH / 256));               // 256 blocks
    const int st = (int)(3 * BH);                      // h ping-pong + c

    for (int layer = 0; layer < 2; ++layer) {
        zero_f32<<<(st + 255) / 256, blk, 0, stream>>>(hbuf, st);

        if (layer == 0)
            gemm_bias_wmma<<<gemmGrid, blk, 0, stream>>>(x, Wih0, bih0, bhh0, xproj, HD);
        else
            gemm_bias_wmma<<<gemmGrid, blk, 0, stream>>>(y0, Wih1, bih1, bhh1, xproj, HD);

        const float* Whh = (layer == 0) ? Whh0 : Whh1;
        float* ys = (layer == 0) ? y0 : y1;

        for (int t = 0; t < SEQL; ++t) {
            float* hp = hbuf + (size_t)(t & 1) * BH;
            float* hx = hbuf + (size_t)((t + 1) & 1) * BH;
            lstm_step_partial<<<stepGrid, blk, 0, stream>>>(Whh, hp, part);
            lstm_gate_reduce<<<gateGrid, blk, 0, stream>>>(
                xproj + (size_t)t * BSZ * N4, part, hx, cbuf, ys + (size_t)t * BH);
        }
        // last step (t=511, odd) wrote parity 0
        copy_f32<<<((int)BH + 255) / 256, blk, 0, stream>>>(hn + (size_t)layer * BH, hbuf, (int)BH);
        copy_f32<<<((int)BH + 255) / 256, blk, 0, stream>>>(cn + (size_t)layer * BH, cbuf, (int)BH);
    }
    (void)in_sizes; (void)n_in; (void)out_size; (void)ws_size;
}